// SSD_42923903156984
// MI455X (gfx1250) — compile-verified
//
#include <hip/hip_runtime.h>
#include <hip/hip_bf16.h>
#include <math.h>

#define N_IMG      4
#define A_TOT      24564
#define C_CLS      80
#define NCLS1      81
#define N_LVL      7
#define PER_IMG    (A_TOT * C_CLS)          // 1,965,120
#define TE         (N_IMG * PER_IMG)        // 7,860,480
#define PAIR_TOT   (N_IMG * N_LVL)          // 28
#define CAND_PER_IMG 6320
#define NMS_PRE_K  2000
#define IMTOP      200
#define CANVAS_F   512.0f
#define NMS_THR_F  0.45f
#define SCORE_THR_F 0.02f

__constant__ int c_lvstart[8] = {0, 16384, 22528, 24064, 24448, 24544, 24560, 24564};
__constant__ int c_lvk[7]     = {1000, 1000, 1000, 1000, 1000, 1000, 320};
__constant__ int c_coff[7]    = {0, 1000, 2000, 3000, 4000, 5000, 6000};

__device__ __forceinline__ int lvl_of(int a) {
  int l = 0;
#pragma unroll
  for (int q = 1; q < N_LVL; ++q) l += (a >= c_lvstart[q]) ? 1 : 0;
  return l;
}

// ---------------- init: zero counters/histograms, seed select state ----------
__global__ void k_init(unsigned* selp, unsigned* cnt, unsigned* histg) {
  int t = blockIdx.x * blockDim.x + threadIdx.x;
  if (t < PAIR_TOT * 256) histg[t] = 0u;
  if (t < PAIR_TOT * 2)   cnt[t]   = 0u;
  if (t < PAIR_TOT) {
    selp[2 * t + 0] = 0u;                              // prefix (threshold bits)
    selp[2 * t + 1] = (unsigned)c_lvk[t % N_LVL];      // remaining k
  }
}

// ---------------- softmax + score threshold ---------------------------------
__global__ void k_softmax(const float* __restrict__ logits, float* __restrict__ scr) {
  int t = blockIdx.x * blockDim.x + threadIdx.x;
  if (t >= N_IMG * A_TOT) return;
  const float* p = logits + (size_t)t * NCLS1;
  float m = p[0];
  for (int c = 1; c < NCLS1; ++c) m = fmaxf(m, p[c]);
  float s = 0.f;
  for (int c = 0; c < NCLS1; ++c) s += expf(p[c] - m);
  float inv = 1.f / s;
  float* o = scr + (size_t)t * C_CLS;
  for (int c = 0; c < C_CLS; ++c) {
    float v = expf(p[c] - m) * inv;       // background (class 80) dropped
    o[c] = (v > SCORE_THR_F) ? v : 0.f;
  }
}

// ---------------- radix-select round: histogram of matching elements --------
__global__ void k_hist(const float* __restrict__ scr, const unsigned* __restrict__ selp,
                       unsigned* __restrict__ histg, int r) {
  __shared__ unsigned sh[256];
  __shared__ int s_pair0, s_uniform;
  int t = blockIdx.x * blockDim.x + threadIdx.x;
  bool valid = (t < TE);
  int pair = -1;
  if (valid) {
    int img = t / PER_IMG;
    int rem = t - img * PER_IMG;
    int a   = rem / C_CLS;
    pair = img * N_LVL + lvl_of(a);
  }
  if (threadIdx.x == 0) { s_pair0 = pair; s_uniform = 1; }
  sh[threadIdx.x] = 0u;
  __syncthreads();
  if (pair != s_pair0) s_uniform = 0;   // benign race: all writers write 0
  __syncthreads();
  int uniform = s_uniform;
  if (valid) {
    unsigned maskhi = (r == 3) ? 0u : (0xFFFFFFFFu << ((r + 1) * 8));
    unsigned prefix = selp[pair * 2];
    unsigned b = ((const unsigned*)scr)[t];
    if ((b & maskhi) == prefix) {
      unsigned bin = (b >> (r * 8)) & 0xFFu;
      if (uniform) atomicAdd(&sh[bin], 1u);
      else         atomicAdd(&histg[pair * 256 + bin], 1u);
    }
  }
  __syncthreads();
  if (uniform && s_pair0 >= 0) {
    unsigned c = sh[threadIdx.x];
    if (c) atomicAdd(&histg[s_pair0 * 256 + threadIdx.x], c);
  }
}

// ---------------- radix-select round: pick byte, update prefix/remaining ----
__global__ void k_pick(unsigned* selp, unsigned* histg, int r) {
  int p = threadIdx.x;
  if (p >= PAIR_TOT) return;
  unsigned rem = selp[p * 2 + 1];
  unsigned prefix = selp[p * 2 + 0];
  unsigned* h = histg + p * 256;
  unsigned cum = 0;
  int q = 255;
  for (; q > 0; --q) {
    unsigned c = h[q];
    if (cum + c >= rem) break;
    cum += c;
  }
  selp[p * 2 + 0] = prefix | ((unsigned)q << (r * 8));
  selp[p * 2 + 1] = rem - cum;
  for (int i = 0; i < 256; ++i) h[i] = 0u;   // reset for next round
}

// ---------------- compaction: emit exactly k candidates per (img,level) -----
__global__ void k_compact(const float* __restrict__ scr, const unsigned* __restrict__ selp,
                          unsigned* __restrict__ cnt, unsigned long long* __restrict__ cand) {
  int t = blockIdx.x * blockDim.x + threadIdx.x;
  if (t >= TE) return;
  int img  = t / PER_IMG;
  int gidx = t - img * PER_IMG;       // a*80 + c within image
  int a    = gidx / C_CLS;
  int lvl  = lvl_of(a);
  int pair = img * N_LVL + lvl;
  unsigned tb  = selp[pair * 2 + 0];
  unsigned neq = selp[pair * 2 + 1];
  unsigned b = ((const unsigned*)scr)[t];
  if (b < tb) return;
  int k = c_lvk[lvl];
  unsigned long long key =
      ((unsigned long long)b << 32) | (unsigned long long)(0xFFFFFFFFu - (unsigned)gidx);
  unsigned long long* dst = cand + (size_t)img * CAND_PER_IMG + c_coff[lvl];
  unsigned gt_cap = (unsigned)k - neq;
  if (b > tb) {
    unsigned pos = atomicAdd(&cnt[pair * 2 + 0], 1u);
    dst[pos] = key;                              // exactly gt_cap of these
  } else {
    unsigned pos = atomicAdd(&cnt[pair * 2 + 1], 1u);
    if (pos < neq) dst[gt_cap + pos] = key;
  }
}

// ---------------- bitonic sort (descending) over u64 keys in LDS ------------
__device__ void bitonic_desc(unsigned long long* s, int n) {
  for (int k2 = 2; k2 <= n; k2 <<= 1) {
    for (int j = k2 >> 1; j > 0; j >>= 1) {
      for (int i = threadIdx.x; i < n; i += blockDim.x) {
        int ixj = i ^ j;
        if (ixj > i) {
          unsigned long long a = s[i], b = s[ixj];
          bool up = ((i & k2) == 0);
          if (up ? (a < b) : (a > b)) { s[i] = b; s[ixj] = a; }
        }
      }
      __syncthreads();
    }
  }
}

// ---------------- per-image: sort 6320 candidates, take top-2000 ------------
__global__ void k_sortcand(const unsigned long long* __restrict__ cand,
                           float* __restrict__ v2, unsigned* __restrict__ anch,
                           unsigned* __restrict__ cls) {
  __shared__ unsigned long long s[8192];
  int img = blockIdx.x;
  for (int i = threadIdx.x; i < 8192; i += blockDim.x)
    s[i] = (i < CAND_PER_IMG) ? cand[(size_t)img * CAND_PER_IMG + i] : 0ull;
  __syncthreads();
  bitonic_desc(s, 8192);
  for (int j = threadIdx.x; j < NMS_PRE_K; j += blockDim.x) {
    unsigned long long k = s[j];
    unsigned vb = (unsigned)(k >> 32);
    unsigned gi = 0xFFFFFFFFu - (unsigned)(k & 0xFFFFFFFFull);
    if (gi >= (unsigned)PER_IMG) gi = 0;
    v2[img * NMS_PRE_K + j]  = __uint_as_float(vb);
    anch[img * NMS_PRE_K + j] = gi / C_CLS;
    cls[img * NMS_PRE_K + j]  = gi % C_CLS;
  }
}

// ---------------- decode boxes + class-offset boxes -------------------------
__global__ void k_decode(const float* __restrict__ priors, const float* __restrict__ reg,
                         const unsigned* __restrict__ anch, const unsigned* __restrict__ cls,
                         float* __restrict__ boxes, float* __restrict__ nb) {
  int t = blockIdx.x * blockDim.x + threadIdx.x;
  if (t >= N_IMG * NMS_PRE_K) return;
  int img = t / NMS_PRE_K;
  unsigned a = anch[t];
  if (a >= (unsigned)A_TOT) a = 0;
  unsigned c = cls[t];
  const float* p = priors + (size_t)a * 4;
  const float* r = reg + ((size_t)img * A_TOT + a) * 4;
  float cx = p[0] + r[0] * 0.1f * p[2];
  float cy = p[1] + r[1] * 0.1f * p[3];
  float w  = p[2] * expf(r[2] * 0.2f);
  float h  = p[3] * expf(r[3] * 0.2f);
  float x0 = fminf(fmaxf(cx - 0.5f * w, 0.f), CANVAS_F);
  float y0 = fminf(fmaxf(cy - 0.5f * h, 0.f), CANVAS_F);
  float x1 = fminf(fmaxf(cx + 0.5f * w, 0.f), CANVAS_F);
  float y1 = fminf(fmaxf(cy + 0.5f * h, 0.f), CANVAS_F);
  float* ob = boxes + (size_t)t * 4;
  ob[0] = x0; ob[1] = y0; ob[2] = x1; ob[3] = y1;
  float off = (float)c * (2.0f * CANVAS_F);
  float* on = nb + (size_t)t * 4;
  on[0] = x0 + off; on[1] = y0 + off; on[2] = x1 + off; on[3] = y1 + off;
}

// ---------------- IoU suppression bit-matrix (async LDS staging) ------------
__global__ void k_ioumask(const float* __restrict__ nb, unsigned long long* __restrict__ mask) {
  __shared__ float s_nb[NMS_PRE_K * 4];       // 32 KB
  int img = blockIdx.y;
  int rb  = blockIdx.x;                       // rows rb*8 .. rb*8+7
  const float* src = nb + (size_t)img * NMS_PRE_K * 4;
  unsigned lds_base = (unsigned)(unsigned long long)(uintptr_t)&s_nb[0];
  // gfx1250 async global->LDS copy engine (ASYNCcnt path)
  for (int v = threadIdx.x; v < NMS_PRE_K; v += blockDim.x) {
    unsigned loff = lds_base + (unsigned)(v * 16);
    unsigned long long gaddr = (unsigned long long)(uintptr_t)(src + (size_t)v * 4);
    asm volatile("global_load_async_to_lds_b128 %0, %1, off"
                 :: "v"(loff), "v"(gaddr) : "memory");
  }
  asm volatile("s_wait_asynccnt 0" ::: "memory");
  __syncthreads();

  int i = rb * 8 + (threadIdx.x >> 5);        // row
  int w = threadIdx.x & 31;                   // 64-bit word within row
  float x0 = s_nb[i * 4 + 0], y0 = s_nb[i * 4 + 1];
  float x1 = s_nb[i * 4 + 2], y1 = s_nb[i * 4 + 3];
  float ai = (x1 - x0) * (y1 - y0);
  unsigned long long bits = 0ull;
  int j0 = w * 64;
#pragma unroll 4
  for (int b = 0; b < 64; ++b) {
    int j = j0 + b;
    if (j > i && j < NMS_PRE_K) {
      float u0 = s_nb[j * 4 + 0], u1 = s_nb[j * 4 + 1];
      float u2 = s_nb[j * 4 + 2], u3 = s_nb[j * 4 + 3];
      float iw = fmaxf(fminf(x1, u2) - fmaxf(x0, u0), 0.f);
      float ih = fmaxf(fminf(y1, u3) - fmaxf(y0, u1), 0.f);
      float inter = iw * ih;
      float aj = (u2 - u0) * (u3 - u1);
      float iou = inter / fmaxf(ai + aj - inter, 1e-6f);
      if (iou > NMS_THR_F) bits |= (1ull << b);
    }
  }
  mask[((size_t)img * NMS_PRE_K + i) * 32 + w] = bits;
}

// ---------------- sequential greedy OR-scan: one wave32 per image -----------
__device__ __forceinline__ unsigned long long shfl64(unsigned long long v, int src) {
  unsigned lo = __shfl((unsigned)(v & 0xFFFFFFFFull), src, 32);
  unsigned hi = __shfl((unsigned)(v >> 32), src, 32);
  return ((unsigned long long)hi << 32) | (unsigned long long)lo;
}

__global__ void k_nmsscan(const unsigned long long* __restrict__ mask,
                          const float* __restrict__ v2, float* __restrict__ fs) {
  int img = blockIdx.x;
  int l = threadIdx.x;                       // lane owns bits [64l, 64l+64)
  unsigned long long acc = 0ull;
  const unsigned long long* mk = mask + (size_t)img * NMS_PRE_K * 32;
  for (int i = 0; i < NMS_PRE_K; ++i) {
    unsigned long long wsup = shfl64(acc, i >> 6);
    bool sup_i = ((wsup >> (i & 63)) & 1ull) != 0ull;
    unsigned long long row = mk[(size_t)i * 32 + l];
    if (!sup_i) acc |= row;
  }
  for (int b = 0; b < 64; ++b) {
    int j = l * 64 + b;
    if (j < NMS_PRE_K) {
      float v = v2[img * NMS_PRE_K + j];
      bool keep = ((acc >> b) & 1ull) == 0ull;
      fs[img * NMS_PRE_K + j] = (keep && v > 0.f) ? v : 0.f;
    }
  }
}

// ---------------- final top-200 + gather outputs ----------------------------
__global__ void k_final(const float* __restrict__ fs, const float* __restrict__ boxes,
                        const unsigned* __restrict__ cls, float* __restrict__ out) {
  __shared__ unsigned long long s[2048];
  int img = blockIdx.x;
  for (int i = threadIdx.x; i < 2048; i += blockDim.x) {
    if (i < NMS_PRE_K) {
      unsigned vb = __float_as_uint(fs[img * NMS_PRE_K + i]);
      s[i] = ((unsigned long long)vb << 32) | (unsigned long long)(0xFFFFFFFFu - (unsigned)i);
    } else s[i] = 0ull;
  }
  __syncthreads();
  bitonic_desc(s, 2048);
  if (threadIdx.x < IMTOP) {
    int t = threadIdx.x;
    unsigned long long k = s[t];
    float v = __uint_as_float((unsigned)(k >> 32));
    unsigned j = 0xFFFFFFFFu - (unsigned)(k & 0xFFFFFFFFull);
    if (j >= (unsigned)NMS_PRE_K) j = 0;
    float b0 = 0.f, b1 = 0.f, b2 = 0.f, b3 = 0.f;
    if (v > 0.f) {
      const float* bb = boxes + ((size_t)img * NMS_PRE_K + j) * 4;
      b0 = bb[0]; b1 = bb[1]; b2 = bb[2]; b3 = bb[3];
    }
    float* fb = out + ((size_t)img * IMTOP + t) * 4;
    fb[0] = b0; fb[1] = b1; fb[2] = b2; fb[3] = b3;
    out[N_IMG * IMTOP * 4 + img * IMTOP + t] = v;
    out[N_IMG * IMTOP * 5 + img * IMTOP + t] = (float)cls[img * NMS_PRE_K + j];
  }
}

// ---------------- host launcher ---------------------------------------------
extern "C" void kernel_launch(void* const* d_in, const int* in_sizes, int n_in,
                              void* d_out, int out_size, void* d_ws, size_t ws_size,
                              hipStream_t stream) {
  (void)in_sizes; (void)n_in; (void)out_size; (void)ws_size;
  const float* cls_logits = (const float*)d_in[0];
  const float* reg        = (const float*)d_in[1];
  const float* priors     = (const float*)d_in[2];
  float* out = (float*)d_out;
  char* ws = (char*)d_ws;
  size_t off = 0;
  auto take = [&](size_t bytes) -> void* {
    void* p = ws + off;
    off += (bytes + 255) & ~(size_t)255;
    return p;
  };
  float*               scr   = (float*)take((size_t)TE * 4);                          // 31.4 MB
  unsigned*            selp  = (unsigned*)take(PAIR_TOT * 2 * 4);
  unsigned*            cnt   = (unsigned*)take(PAIR_TOT * 2 * 4);
  unsigned*            histg = (unsigned*)take(PAIR_TOT * 256 * 4);
  unsigned long long*  cand  = (unsigned long long*)take((size_t)N_IMG * CAND_PER_IMG * 8);
  float*               v2    = (float*)take((size_t)N_IMG * NMS_PRE_K * 4);
  unsigned*            anch  = (unsigned*)take((size_t)N_IMG * NMS_PRE_K * 4);
  unsigned*            clsb  = (unsigned*)take((size_t)N_IMG * NMS_PRE_K * 4);
  float*               boxes = (float*)take((size_t)N_IMG * NMS_PRE_K * 4 * 4);
  float*               nb    = (float*)take((size_t)N_IMG * NMS_PRE_K * 4 * 4);
  unsigned long long*  mask  = (unsigned long long*)take((size_t)N_IMG * NMS_PRE_K * 32 * 8); // 2 MB
  float*               fs    = (float*)take((size_t)N_IMG * NMS_PRE_K * 4);

  k_init<<<dim3((PAIR_TOT * 256 + 255) / 256), 256, 0, stream>>>(selp, cnt, histg);
  k_softmax<<<dim3((N_IMG * A_TOT + 255) / 256), 256, 0, stream>>>(cls_logits, scr);
  for (int r = 3; r >= 0; --r) {
    k_hist<<<dim3((TE + 255) / 256), 256, 0, stream>>>(scr, selp, histg, r);
    k_pick<<<1, 32, 0, stream>>>(selp, histg, r);
  }
  k_compact<<<dim3((TE + 255) / 256), 256, 0, stream>>>(scr, selp, cnt, cand);
  k_sortcand<<<N_IMG, 1024, 0, stream>>>(cand, v2, anch, clsb);
  k_decode<<<dim3((N_IMG * NMS_PRE_K + 255) / 256), 256, 0, stream>>>(priors, reg, anch, clsb,
                                                                      boxes, nb);
  k_ioumask<<<dim3(NMS_PRE_K / 8, N_IMG), 256, 0, stream>>>(nb, mask);
  k_nmsscan<<<N_IMG, 32, 0, stream>>>(mask, v2, fs);
  k_final<<<N_IMG, 1024, 0, stream>>>(fs, boxes, clsb, out);
}